// VOX_mean_89558658056497
// MI455X (gfx1250) — compile-verified
//
#include <hip/hip_runtime.h>

#define BB 8
#define NN 65536
#define GRID3 729
#define NVOX (BB * GRID3)      // 5832
#define CRN (NVOX * 64)        // 373248

// flip to 0 to revert to the register-staged copy path
#define TRY_ASYNC_ASM 1

typedef float v2f __attribute__((ext_vector_type(2)));
typedef float v8f __attribute__((ext_vector_type(8)));

#if defined(__has_builtin)
#  if __has_builtin(__builtin_amdgcn_wmma_f32_16x16x4_f32)
#    define USE_WMMA_F32X4 1
#  endif
#endif

#if TRY_ASYNC_ASM && defined(__gfx1250__)
#  define USE_ASYNC_COPY 1
#endif

// D(16x16,f32) = A(16x4,f32) * B(4x16,f32) + C   (one wave32)
__device__ __forceinline__ v8f wmma4(v2f a, v2f b, v8f c) {
#ifdef USE_WMMA_F32X4
  return __builtin_amdgcn_wmma_f32_16x16x4_f32(false, a, false, b, (short)0, c,
                                               false, false);
#else
  // exact lane-semantics fallback via cross-lane shuffles (only if builtin absent)
  const int lane = (int)(threadIdx.x & 31u);
  const int lm = lane & 15;
  const int hi = (lane >> 4) & 1;
  const float b0 = __shfl(b.x, lm);
  const float b1 = __shfl(b.y, lm);
  const float b2 = __shfl(b.x, lm + 16);
  const float b3 = __shfl(b.y, lm + 16);
#pragma unroll
  for (int r = 0; r < 8; ++r) {
    const int m = r + 8 * hi;
    const float a0 = __shfl(a.x, m);
    const float a1 = __shfl(a.y, m);
    const float a2 = __shfl(a.x, m + 16);
    const float a3 = __shfl(a.y, m + 16);
    c[r] += a0 * b0 + a1 * b1 + a2 * b2 + a3 * b3;
  }
  return c;
#endif
}

// 16B global -> LDS copy; async (ASYNCcnt-tracked, no VGPR roundtrip) if available.
// VDST operand = wave-relative LDS byte offset (low 32 bits of the generic addr,
// per ISA flat->LDS mapping); VADDR = 64-bit global address; off = no SADDR.
__device__ __forceinline__ void cp16_g2l(float* lds_dst, const float* gsrc) {
#ifdef USE_ASYNC_COPY
  const unsigned int loff = (unsigned int)(unsigned long long)lds_dst;
  const unsigned long long ga = (unsigned long long)gsrc;
  asm volatile("global_load_async_to_lds_b128 %0, %1, off"
               :
               : "v"(loff), "v"(ga)
               : "memory");
#else
  *(float4*)lds_dst = *(const float4*)gsrc;
#endif
}

__device__ __forceinline__ void wait_async_copies() {
#ifdef USE_ASYNC_COPY
#  if defined(__has_builtin) && __has_builtin(__builtin_amdgcn_s_wait_asynccnt)
  __builtin_amdgcn_s_wait_asynccnt(0);
#  else
  asm volatile("s_wait_asynccnt 0x0" ::: "memory");
#  endif
#endif
}

__device__ __forceinline__ void wait_lds_reads_done() {
#ifdef USE_ASYNC_COPY
  asm volatile("s_wait_dscnt 0x0" ::: "memory");
#endif
}

__device__ __forceinline__ int wrap9(int v) {
  return v < 0 ? v + 9 : (v >= 9 ? v - 9 : v);
}

// ---------------- zero scratch ----------------
__global__ __launch_bounds__(256) void k_zero(float* __restrict__ CR,
                                              float* __restrict__ p1,
                                              float* __restrict__ cs) {
  const int idx = blockIdx.x * 256 + threadIdx.x;
  if (idx < CRN) CR[idx] = 0.f;
  if (idx < 8192) p1[idx] = 0.f;
  if (idx < 512) cs[idx] = 0.f;
}

// ---------------- pass A: scatter + softmax + pooled + colsum ----------------
__global__ __launch_bounds__(128) void k_stats(
    const float* __restrict__ x, const int* __restrict__ vox,
    const float* __restrict__ aw, float* __restrict__ CR,
    float* __restrict__ pooled1, float* __restrict__ colsum) {
  __shared__ float s_aw[64][20];    // attn_weights [k][l]
  __shared__ float s_x[4][16][68];  // per-wave x tile [point][k]
  __shared__ float s_A[4][16][18];  // per-wave softmax tile [point][l]

  const int tid = threadIdx.x;
  const int w = tid >> 5, lane = tid & 31;
  const int hi = lane >> 4, lm = lane & 15;
  const int b = blockIdx.x >> 6, blk = blockIdx.x & 63;

  for (int i = tid; i < 64 * 16; i += 128) s_aw[i >> 4][i & 15] = aw[i];
  __syncthreads();

  v8f pacc[4];
#pragma unroll
  for (int j = 0; j < 4; ++j) pacc[j] = (v8f){0.f, 0.f, 0.f, 0.f, 0.f, 0.f, 0.f, 0.f};
  float cs0 = 0.f, cs1 = 0.f;

  for (int t = 0; t < 16; ++t) {
    const int p0 = b * NN + blk * 1024 + w * 256 + t * 16;
    // load x tile -> LDS, scatter-add to conv_ready (x needed in VGPRs anyway)
#pragma unroll
    for (int i = 0; i < 8; ++i) {
      const int q = lane + 32 * i;
      const int row = q >> 4, c4 = (q & 15) << 2;
      const float4 f = *(const float4*)(x + (size_t)(p0 + row) * 64 + c4);
      const int v = vox[p0 + row];
      float* crp = CR + (size_t)v * 64 + c4;
      atomicAdd(crp + 0, f.x);
      atomicAdd(crp + 1, f.y);
      atomicAdd(crp + 2, f.z);
      atomicAdd(crp + 3, f.w);
      *(float4*)&s_x[w][row][c4] = f;
    }
    // colsum partials: lane owns channels (2*lane, 2*lane+1)
#pragma unroll
    for (int row = 0; row < 16; ++row) {
      const float2 g = *(const float2*)&s_x[w][row][2 * lane];
      cs0 += g.x;
      cs1 += g.y;
    }
    // w' = x(16x64) @ attn_w(64x16) : 16 WMMAs
    v8f wp = (v8f){0.f, 0.f, 0.f, 0.f, 0.f, 0.f, 0.f, 0.f};
#pragma unroll
    for (int k0 = 0; k0 < 64; k0 += 4) {
      v2f a, bb;
      a.x = s_x[w][lm][k0 + 2 * hi];
      a.y = s_x[w][lm][k0 + 2 * hi + 1];
      bb.x = s_aw[k0 + 2 * hi][lm];
      bb.y = s_aw[k0 + 2 * hi + 1][lm];
      wp = wmma4(a, bb, wp);
    }
    // softmax per row (row's 16 logits live across 16 lanes of a half)
#pragma unroll
    for (int r = 0; r < 8; ++r) {
      const float v = wp[r];
      float mx = v;
      mx = fmaxf(mx, __shfl_xor(mx, 1));
      mx = fmaxf(mx, __shfl_xor(mx, 2));
      mx = fmaxf(mx, __shfl_xor(mx, 4));
      mx = fmaxf(mx, __shfl_xor(mx, 8));
      const float e = __expf(v - mx);
      float s = e;
      s += __shfl_xor(s, 1);
      s += __shfl_xor(s, 2);
      s += __shfl_xor(s, 4);
      s += __shfl_xor(s, 8);
      s_A[w][r + 8 * hi][lm] = e / s;
    }
    // pooled += A^T(16x16) @ x(16x64) : 16 WMMAs (M=class, K=point, N=channel)
#pragma unroll
    for (int k0 = 0; k0 < 16; k0 += 4) {
      v2f a;
      a.x = s_A[w][k0 + 2 * hi][lm];
      a.y = s_A[w][k0 + 2 * hi + 1][lm];
#pragma unroll
      for (int j = 0; j < 4; ++j) {
        v2f bb;
        bb.x = s_x[w][k0 + 2 * hi][16 * j + lm];
        bb.y = s_x[w][k0 + 2 * hi + 1][16 * j + lm];
        pacc[j] = wmma4(a, bb, pacc[j]);
      }
    }
  }
  atomicAdd(&colsum[b * 64 + 2 * lane], cs0);
  atomicAdd(&colsum[b * 64 + 2 * lane + 1], cs1);
#pragma unroll
  for (int j = 0; j < 4; ++j)
#pragma unroll
    for (int r = 0; r < 8; ++r)
      atomicAdd(&pooled1[b * 1024 + (r + 8 * hi) * 64 + 16 * j + lm], pacc[j][r]);
}

// ---------------- tiny: everything + pooled2 ----------------
__global__ __launch_bounds__(256) void k_small(
    const float* __restrict__ colsum, const float* __restrict__ p1,
    const float* __restrict__ lw, const float* __restrict__ cw,
    float* __restrict__ every, float* __restrict__ p2) {
  const int idx = blockIdx.x * 256 + threadIdx.x;
  if (idx < 512) {
    const int b = idx >> 6, o = idx & 63;
    float s = 0.f;
    for (int k = 0; k < 64; ++k) s += colsum[b * 64 + k] * lw[o * 64 + k];
    every[idx] = s * (1.0f / 65536.0f);
  } else if (idx < 512 + 8192) {
    const int t = idx - 512;
    const int b = t >> 10, h = (t >> 6) & 15, o = t & 63;
    float s = 0.f;
    for (int k = 0; k < 64; ++k) {
      const float* pr = p1 + b * 1024 + k;                     // stride 64 over l
      const float* cr = cw + (size_t)k * 16384 + o * 256 + h;  // stride 16 over l
#pragma unroll
      for (int l = 0; l < 16; ++l) s += pr[l * 64] * cr[l * 16];
    }
    p2[t] = s;  // [b][h][o]
  }
}

// ---------------- in-place mean normalize (raw-view semantics) ----------------
__global__ __launch_bounds__(256) void k_mean(float* __restrict__ CR,
                                              const float* __restrict__ np) {
  const int idx = blockIdx.x * 256 + threadIdx.x;
  if (idx < CRN) CR[idx] = CR[idx] / np[idx >> 6];
}

// ---------------- 3D wrap conv, output flat [b][o][729] ----------------
__global__ __launch_bounds__(256) void k_conv(const float* __restrict__ mb,
                                              const float* __restrict__ cwp,
                                              const float* __restrict__ cb,
                                              float* __restrict__ outc) {
  const int idx = blockIdx.x * 256 + threadIdx.x;
  if (idx >= BB * 64 * 81) return;
  const int dh = idx % 81;
  const int bo = idx / 81;
  const int o = bo & 63, b = bo >> 6;
  const int d = dh / 9, h = dh % 9;
  const float bias = cb[o];
  float acc[9];
#pragma unroll
  for (int xx = 0; xx < 9; ++xx) acc[xx] = bias;
  const float* m0 = mb + (size_t)b * 46656;
  const float* w0 = cwp + (size_t)o * 1728;
  for (int i = 0; i < 64; ++i) {
#pragma unroll
    for (int dz = 0; dz < 3; ++dz) {
      const int zd = wrap9(d - 1 + dz);
#pragma unroll
      for (int dy = 0; dy < 3; ++dy) {
        const int yd = wrap9(h - 1 + dy);
        const float* row = m0 + i * 729 + zd * 81 + yd * 9;
        float g[9];
#pragma unroll
        for (int xx = 0; xx < 9; ++xx) g[xx] = row[xx];
        const float* wv = w0 + i * 27 + dz * 9 + dy * 3;
        const float wa = wv[0], wb = wv[1], wc = wv[2];
#pragma unroll
        for (int xx = 0; xx < 9; ++xx)
          acc[xx] += g[(xx + 8) % 9] * wa + g[xx] * wb + g[(xx + 1) % 9] * wc;
      }
    }
  }
  float* op = outc + (size_t)bo * 729 + d * 81 + h * 9;
#pragma unroll
  for (int xx = 0; xx < 9; ++xx) op[xx] = acc[xx];
}

// ---------------- pass B: attn + gamma WMMA chain + gather epilogue ----------------
__global__ __launch_bounds__(128) void k_final(
    const float* __restrict__ x, const int* __restrict__ vox,
    const float* __restrict__ aw, const float* __restrict__ gw,
    const float* __restrict__ gb, const float* __restrict__ p2,
    const float* __restrict__ every, const float* __restrict__ convo,
    float* __restrict__ out) {
  __shared__ float s_gT[64][66];    // gamma_w^T [k][o]
  __shared__ float s_aw[64][20];    // attn_weights [k][l]
  __shared__ float s_p2[16][68];    // pooled2[b] [l][o]
  __shared__ float s_x[4][16][68];  // per-wave x tile
  __shared__ float s_A[4][16][18];  // per-wave softmax tile

  const int tid = threadIdx.x;
  const int w = tid >> 5, lane = tid & 31;
  const int hi = lane >> 4, lm = lane & 15;
  const int b = blockIdx.x >> 7, blk = blockIdx.x & 127;

  for (int i = tid; i < 64 * 64; i += 128) {
    const int o = i >> 6, k = i & 63;
    s_gT[k][o] = gw[i];
  }
  for (int i = tid; i < 64 * 16; i += 128) s_aw[i >> 4][i & 15] = aw[i];
  for (int i = tid; i < 16 * 64; i += 128) s_p2[i >> 6][i & 63] = p2[b * 1024 + i];
  __syncthreads();

  for (int t = 0; t < 8; ++t) {
    const int p0 = b * NN + blk * 512 + w * 128 + t * 16;
    // stage x tile into LDS (async global->LDS; enforce WAR vs previous tile's
    // DS reads explicitly since ASYNCcnt and DScnt are independent queues)
    wait_lds_reads_done();
#pragma unroll
    for (int i = 0; i < 8; ++i) {
      const int q = lane + 32 * i;
      const int row = q >> 4, c4 = (q & 15) << 2;
      cp16_g2l(&s_x[w][row][c4], x + (size_t)(p0 + row) * 64 + c4);
    }
    if (t + 1 < 8) __builtin_prefetch(x + (size_t)(p0 + 16) * 64, 0, 1);
    wait_async_copies();

    // w' = x @ attn_w
    v8f wp = (v8f){0.f, 0.f, 0.f, 0.f, 0.f, 0.f, 0.f, 0.f};
#pragma unroll
    for (int k0 = 0; k0 < 64; k0 += 4) {
      v2f a, bb;
      a.x = s_x[w][lm][k0 + 2 * hi];
      a.y = s_x[w][lm][k0 + 2 * hi + 1];
      bb.x = s_aw[k0 + 2 * hi][lm];
      bb.y = s_aw[k0 + 2 * hi + 1][lm];
      wp = wmma4(a, bb, wp);
    }
    // softmax -> s_A[point][l]
#pragma unroll
    for (int r = 0; r < 8; ++r) {
      const float v = wp[r];
      float mx = v;
      mx = fmaxf(mx, __shfl_xor(mx, 1));
      mx = fmaxf(mx, __shfl_xor(mx, 2));
      mx = fmaxf(mx, __shfl_xor(mx, 4));
      mx = fmaxf(mx, __shfl_xor(mx, 8));
      const float e = __expf(v - mx);
      float s = e;
      s += __shfl_xor(s, 1);
      s += __shfl_xor(s, 2);
      s += __shfl_xor(s, 4);
      s += __shfl_xor(s, 8);
      s_A[w][r + 8 * hi][lm] = e / s;
    }

    v8f acc[4];
#pragma unroll
    for (int j = 0; j < 4; ++j) acc[j] = (v8f){0.f, 0.f, 0.f, 0.f, 0.f, 0.f, 0.f, 0.f};

    // attn_out = A(16x16) @ pooled2(16x64)
#pragma unroll
    for (int k0 = 0; k0 < 16; k0 += 4) {
      v2f a;
      a.x = s_A[w][lm][k0 + 2 * hi];
      a.y = s_A[w][lm][k0 + 2 * hi + 1];
#pragma unroll
      for (int j = 0; j < 4; ++j) {
        v2f bb;
        bb.x = s_p2[k0 + 2 * hi][16 * j + lm];
        bb.y = s_p2[k0 + 2 * hi + 1][16 * j + lm];
        acc[j] = wmma4(a, bb, acc[j]);
      }
    }
    // += x(16x64) @ gamma_w^T(64x64)
#pragma unroll
    for (int k0 = 0; k0 < 64; k0 += 4) {
      v2f a;
      a.x = s_x[w][lm][k0 + 2 * hi];
      a.y = s_x[w][lm][k0 + 2 * hi + 1];
#pragma unroll
      for (int j = 0; j < 4; ++j) {
        v2f bb;
        bb.x = s_gT[k0 + 2 * hi][16 * j + lm];
        bb.y = s_gT[k0 + 2 * hi + 1][16 * j + lm];
        acc[j] = wmma4(a, bb, acc[j]);
      }
    }
    // epilogue: + gather(conv raw view) + everything + gamma_b
    int vr[8];
#pragma unroll
    for (int r = 0; r < 8; ++r) vr[r] = vox[p0 + r + 8 * hi];
#pragma unroll
    for (int j = 0; j < 4; ++j) {
      const int oc = 16 * j + lm;
      const float ev = every[b * 64 + oc] + gb[oc];
#pragma unroll
      for (int r = 0; r < 8; ++r) {
        const int p = p0 + r + 8 * hi;
        out[(size_t)p * 64 + oc] =
            acc[j][r] + ev + convo[(size_t)vr[r] * 64 + oc];
      }
    }
  }
}

extern "C" void kernel_launch(void* const* d_in, const int* in_sizes, int n_in,
                              void* d_out, int out_size, void* d_ws,
                              size_t ws_size, hipStream_t stream) {
  const float* x = (const float*)d_in[0];
  const int* vox = (const int*)d_in[1];
  const float* np = (const float*)d_in[2];
  const float* aw = (const float*)d_in[3];
  const float* cw = (const float*)d_in[4];
  const float* gw = (const float*)d_in[5];
  const float* gb = (const float*)d_in[6];
  const float* lw = (const float*)d_in[7];
  const float* convw = (const float*)d_in[8];
  const float* convb = (const float*)d_in[9];
  float* out = (float*)d_out;

  float* ws = (float*)d_ws;
  float* CR = ws;                // 373248 (conv_ready, later mean in-place)
  float* convo = ws + 373248;    // 373248 (conv output, flat [b][o][729])
  float* pooled1 = ws + 746496;  // 8192
  float* pooled2 = ws + 754688;  // 8192
  float* colsum = ws + 762880;   // 512
  float* every = ws + 763392;    // 512

  k_zero<<<1459, 256, 0, stream>>>(CR, pooled1, colsum);
  k_stats<<<512, 128, 0, stream>>>(x, vox, aw, CR, pooled1, colsum);
  k_small<<<34, 256, 0, stream>>>(colsum, pooled1, lw, cw, every, pooled2);
  k_mean<<<1459, 256, 0, stream>>>(CR, np);
  k_conv<<<162, 256, 0, stream>>>(CR, convw, convb, convo);
  k_final<<<1024, 128, 0, stream>>>(x, vox, aw, gw, gb, pooled2, every, convo,
                                    out);
}